// LocalInteraction_44332652429560
// MI455X (gfx1250) — compile-verified
//
#include <hip/hip_runtime.h>
#include <hip/hip_bf16.h>
#include <math.h>
#include <stdint.h>

typedef __attribute__((ext_vector_type(2))) float v2f;
typedef __attribute__((ext_vector_type(8))) float v8f;

#define PI_F     3.14159265358979323846f
#define CUT_F    5.0f
#define LOG2_F   0.69314718055994530942f

// ---------------------------------------------------------------------------
// gfx1250 async global->LDS copy (ASYNCcnt-tracked), GV addressing mode.
// vdst = per-lane LDS byte address, 64-bit global address in a VGPR pair.
// ---------------------------------------------------------------------------
__device__ __forceinline__ void async_b128(uint32_t lds_addr, const void* gaddr) {
    asm volatile("global_load_async_to_lds_b128 %0, %1, off"
                 :: "v"(lds_addr), "v"(gaddr)
                 : "memory");
}
__device__ __forceinline__ void wait_asynccnt_le4() {
    asm volatile("s_wait_asynccnt 0x4" ::: "memory");
}
__device__ __forceinline__ void wait_asynccnt_0() {
    asm volatile("s_wait_asynccnt 0x0" ::: "memory");
}

// ---------------------------------------------------------------------------
// Generic: one wave computes a 16x16 tile of (A[16x128] @ B[128x128]) with
// V_WMMA_F32_16X16X4_F32.  A comes from LDS (row stride lds_stride floats,
// padded to kill bank conflicts), B from global (row-major, ldb=128).
// ---------------------------------------------------------------------------
__device__ __forceinline__ v8f wmma_rowtile_f32(const float* __restrict__ Alds,
                                                int lds_stride,
                                                const float* __restrict__ Bglob,
                                                int ldb, int ntile, int lane,
                                                v8f acc) {
    const int m  = lane & 15;
    const int hi = lane >> 4;
    const int n  = ntile * 16 + (lane & 15);
#pragma unroll
    for (int kk = 0; kk < 32; ++kk) {
        const int k0 = kk * 4 + 2 * hi;
        v2f av, bv;
        av.x = Alds[m * lds_stride + k0];
        av.y = Alds[m * lds_stride + k0 + 1];
        bv.x = Bglob[(size_t)k0 * ldb + n];
        bv.y = Bglob[(size_t)(k0 + 1) * ldb + n];
        acc = __builtin_amdgcn_wmma_f32_16x16x4_f32(false, av, false, bv,
                                                    (short)0, acc, false, false);
    }
    return acc;
}

// ---------------------------------------------------------------------------
// Kernel 1: y[8192,128] = x[8192,128] @ W_in2f[128,128]
// ---------------------------------------------------------------------------
__global__ void __launch_bounds__(256)
k_in2f(const float* __restrict__ x, const float* __restrict__ Win2f,
       float* __restrict__ yws) {
    __shared__ float xL[16 * 132];
    const int tid  = threadIdx.x;
    const int row0 = blockIdx.x * 16;
    for (int i = tid; i < 16 * 128; i += 256) {
        const int r = i >> 7, c = i & 127;
        xL[r * 132 + c] = x[(size_t)(row0 + r) * 128 + c];
    }
    __syncthreads();
    const int wave = tid >> 5, lane = tid & 31;
    v8f acc = {};
    acc = wmma_rowtile_f32(xL, 132, Win2f, 128, wave, lane, acc);
    const int hi  = lane >> 4;
    const int col = wave * 16 + (lane & 15);
#pragma unroll
    for (int i = 0; i < 8; ++i) {
        const int m = i + 8 * hi;
        yws[(size_t)(row0 + m) * 128 + col] = acc[i];
    }
}

// ---------------------------------------------------------------------------
// Kernel 2: per (b,a) pair:
//   Wm[64,128] = fs[64,64]@Ws[64,128] + fp[64,64]@Wp[64,128]   (f32 WMMA)
//   sc[n]      = mask * 0.5*(cos(r*pi/cut)+1) * (r<cut)
//   agg[f]     = sum_n sc[n] * y[b, nbr[n], f] * Wm[n,f]       (LDS atomics)
// fs/fp staged via async global->LDS; fp copy overlapped with the fs pass.
// ---------------------------------------------------------------------------
__global__ void __launch_bounds__(256)
k_filterconv(const float* __restrict__ r_ij, const int* __restrict__ neighbors,
             const float* __restrict__ mask, const float* __restrict__ fs,
             const float* __restrict__ fp, const float* __restrict__ Ws,
             const float* __restrict__ Wp, const float* __restrict__ yws,
             float* __restrict__ aggws) {
    __shared__ __align__(16) float fsL[64 * 68];   // row stride 68 f = 272 B
    __shared__ __align__(16) float fpL[64 * 68];
    __shared__ float sc[64];
    __shared__ int   nbr[64];
    __shared__ float aggL[128];

    const int tid  = threadIdx.x;
    const int pair = blockIdx.x;      // = b*1024 + a
    const int b    = pair >> 10;
    const size_t base = (size_t)pair * (64 * 64);

    // ---- issue async copies: 64x64 f32 tile = 1024 x 16B chunks each ----
    {
        const uint32_t fsOff = (uint32_t)(uintptr_t)&fsL[0];
        const uint32_t fpOff = (uint32_t)(uintptr_t)&fpL[0];
        const char* gfs = (const char*)(fs + base);
        const char* gfp = (const char*)(fp + base);
#pragma unroll
        for (int i = 0; i < 4; ++i) {               // 4 instrs/wave -> fs
            const int c = i * 256 + tid;            // chunk id
            const uint32_t l = (uint32_t)((c >> 4) * 272 + (c & 15) * 16);
            async_b128(fsOff + l, gfs + (size_t)c * 16);
        }
#pragma unroll
        for (int i = 0; i < 4; ++i) {               // 4 instrs/wave -> fp
            const int c = i * 256 + tid;
            const uint32_t l = (uint32_t)((c >> 4) * 272 + (c & 15) * 16);
            async_b128(fpOff + l, gfp + (size_t)c * 16);
        }
    }

    if (tid < 64) {
        const float r  = r_ij[(size_t)pair * 64 + tid];
        const float mk = mask[(size_t)pair * 64 + tid];
        const float cc = 0.5f * (cosf(r * (PI_F / CUT_F)) + 1.0f) *
                         (r < CUT_F ? 1.0f : 0.0f);
        sc[tid]  = mk * cc;
        nbr[tid] = neighbors[(size_t)pair * 64 + tid];
    }
    if (tid < 128) aggL[tid] = 0.0f;

    wait_asynccnt_le4();      // fs chunks landed (async completes in order)
    __syncthreads();          // every wave's fs copy + sc/nbr/agg init done

    const int wave  = tid >> 5, lane = tid & 31;
    const int mtile = wave & 3;            // 4 tiles along n (64 rows)
    const int ngrp  = (wave >> 2) * 4;     // 4 tiles along f per wave
    const int m     = mtile * 16 + (lane & 15);
    const int hi    = lane >> 4;

    v8f acc[4] = {{}, {}, {}, {}};

    // ---- pass 0: fs @ Ws (fp still streaming in the background) ----
#pragma unroll
    for (int kk = 0; kk < 16; ++kk) {
        const int k0 = kk * 4 + 2 * hi;
        v2f av;
        av.x = fsL[m * 68 + k0];
        av.y = fsL[m * 68 + k0 + 1];
#pragma unroll
        for (int j = 0; j < 4; ++j) {
            const int n = (ngrp + j) * 16 + (lane & 15);
            v2f bv;
            bv.x = Ws[(size_t)k0 * 128 + n];
            bv.y = Ws[(size_t)(k0 + 1) * 128 + n];
            acc[j] = __builtin_amdgcn_wmma_f32_16x16x4_f32(
                false, av, false, bv, (short)0, acc[j], false, false);
        }
    }

    wait_asynccnt_0();        // fp chunks landed
    __syncthreads();          // all waves done waiting -> fpL fully valid

    // ---- pass 1: += fp @ Wp ----
#pragma unroll
    for (int kk = 0; kk < 16; ++kk) {
        const int k0 = kk * 4 + 2 * hi;
        v2f av;
        av.x = fpL[m * 68 + k0];
        av.y = fpL[m * 68 + k0 + 1];
#pragma unroll
        for (int j = 0; j < 4; ++j) {
            const int n = (ngrp + j) * 16 + (lane & 15);
            v2f bv;
            bv.x = Wp[(size_t)k0 * 128 + n];
            bv.y = Wp[(size_t)(k0 + 1) * 128 + n];
            acc[j] = __builtin_amdgcn_wmma_f32_16x16x4_f32(
                false, av, false, bv, (short)0, acc[j], false, false);
        }
    }

    // ---- gather y over neighbors, scale by sc[n], reduce into aggL[f] ----
#pragma unroll
    for (int j = 0; j < 4; ++j) {
        const int f = (ngrp + j) * 16 + (lane & 15);
        float part = 0.0f;
#pragma unroll
        for (int i = 0; i < 8; ++i) {
            const int n  = mtile * 16 + i + 8 * hi;
            const int nb = nbr[n];
            const float yv = yws[((size_t)b * 1024 + nb) * 128 + f];
            part += sc[n] * yv * acc[j][i];
        }
        atomicAdd(&aggL[f], part);
    }
    __syncthreads();
    if (tid < 128) aggws[(size_t)pair * 128 + tid] = aggL[tid];
}

// ---------------------------------------------------------------------------
// Kernel 3: v = ssp(agg @ W_f2out + b_f2out);  out = v @ W_dense + b_dense
// ---------------------------------------------------------------------------
__global__ void __launch_bounds__(256)
k_out(const float* __restrict__ aggws, const float* __restrict__ Wf2out,
      const float* __restrict__ bf2out, const float* __restrict__ Wdense,
      const float* __restrict__ bdense, float* __restrict__ out) {
    __shared__ float aL[16 * 132];
    __shared__ float vL[16 * 132];
    const int tid  = threadIdx.x;
    const int row0 = blockIdx.x * 16;
    for (int i = tid; i < 16 * 128; i += 256) {
        const int r = i >> 7, c = i & 127;
        aL[r * 132 + c] = aggws[(size_t)(row0 + r) * 128 + c];
    }
    __syncthreads();
    const int wave = tid >> 5, lane = tid & 31;
    const int hi   = lane >> 4;
    const int col  = wave * 16 + (lane & 15);

    v8f acc = {};
    acc = wmma_rowtile_f32(aL, 132, Wf2out, 128, wave, lane, acc);
    const float bb = bf2out[col];
#pragma unroll
    for (int i = 0; i < 8; ++i) {
        const int mrow = i + 8 * hi;
        const float t  = acc[i] + bb;
        const float sp = (t > 20.0f) ? t : log1pf(expf(t));
        vL[mrow * 132 + col] = sp - LOG2_F;
    }
    __syncthreads();

    v8f acc2 = {};
    acc2 = wmma_rowtile_f32(vL, 132, Wdense, 128, wave, lane, acc2);
    const float bd = bdense[col];
#pragma unroll
    for (int i = 0; i < 8; ++i) {
        const int mrow = i + 8 * hi;
        out[(size_t)(row0 + mrow) * 128 + col] = acc2[i] + bd;
    }
}

// ---------------------------------------------------------------------------
extern "C" void kernel_launch(void* const* d_in, const int* in_sizes, int n_in,
                              void* d_out, int out_size, void* d_ws,
                              size_t ws_size, hipStream_t stream) {
    const float* x         = (const float*)d_in[0];
    const float* r_ij      = (const float*)d_in[1];
    const int*   neighbors = (const int*)d_in[2];
    const float* mask      = (const float*)d_in[3];
    const float* fsb       = (const float*)d_in[4];
    const float* fpb       = (const float*)d_in[5];
    const float* Ws        = (const float*)d_in[6];
    const float* Wp        = (const float*)d_in[7];
    const float* Win2f     = (const float*)d_in[8];
    const float* Wf2out    = (const float*)d_in[9];
    const float* bf2out    = (const float*)d_in[10];
    const float* Wdense    = (const float*)d_in[11];
    const float* bdense    = (const float*)d_in[12];
    float* out = (float*)d_out;

    float* yws   = (float*)d_ws;                  // [8192,128]
    float* aggws = yws + (size_t)8192 * 128;      // [8192,128]

    k_in2f<<<512, 256, 0, stream>>>(x, Win2f, yws);
    k_filterconv<<<8192, 256, 0, stream>>>(r_ij, neighbors, mask, fsb, fpb, Ws,
                                           Wp, yws, aggws);
    k_out<<<512, 256, 0, stream>>>(aggws, Wf2out, bf2out, Wdense, bdense, out);
}